// LeapfrogIntegrator_77704548319438
// MI455X (gfx1250) — compile-verified
//
#include <hip/hip_runtime.h>
#include <math.h>

// ---------------------------------------------------------------------------
// Riemannian leapfrog on 32768 independent 16x16 SPD matrices (gfx1250).
// eigh-free reformulation: Newton-Schulz matrix sqrt + scaling-and-squaring
// expm, everything expressed as 16x16x16 f32 matmuls on V_WMMA_F32_16X16X4_F32.
// One wave32 per matrix; LDS tile staging for WMMA A/B/C layout conversion.
// Inputs staged via async global->LDS (ASYNCcnt); B operands read transposed
// (all are symmetric) so both A and B LDS reads merge into 2addr_b64 pairs.
// ---------------------------------------------------------------------------

typedef __attribute__((ext_vector_type(2))) float v2f;
typedef __attribute__((ext_vector_type(8))) float v8f;

#define DT_C 0.01f
#define H_C  0.005f

constexpr int BATCH = 8;
constexpr int NPTS  = 4096;
constexpr int KDIM  = 16;
constexpr int NMAT  = BATCH * NPTS;   // 32768 matrices
constexpr int TILE  = 256;            // floats per 16x16 tile
constexpr int NW    = 4;              // waves per block
constexpr int NTILES = 9;             // LDS tiles per wave

// output layout (floats), in reference return order
constexpr size_t OFF_MU    = 0;
constexpr size_t OFF_SIGMA = OFF_MU    + (size_t)NMAT * KDIM;
constexpr size_t OFF_PHI   = OFF_SIGMA + (size_t)NMAT * KDIM * KDIM;
constexpr size_t OFF_PIMU  = OFF_PHI   + (size_t)NMAT * 3;
constexpr size_t OFF_PISIG = OFF_PIMU  + (size_t)NMAT * KDIM;
constexpr size_t OFF_PIPHI = OFF_PISIG + (size_t)NMAT * KDIM * KDIM;

// Compiler-only ordering barrier. Hardware LDS ops from one wave execute
// in order (ISA 7.3), so same-wave cross-lane store->load needs no s_wait;
// the backend still inserts the load-data->VGPR waits it can see.
__device__ __forceinline__ void lds_fence() {
    asm volatile("" ::: "memory");
}

__device__ __forceinline__ void async_fence() {
    asm volatile("s_wait_asynccnt 0x0" ::: "memory");
}

__device__ __forceinline__ int trI(int idx) {   // transpose index in 16x16 tile
    return ((idx & 15) << 4) | (idx >> 4);
}

// Async global->LDS copy of one 1KB tile (256 floats), 16B per lane x2.
// GVS mode: mem = SADDR + VADDR + inst_offset; LDS dst = VDST + inst_offset.
__device__ __forceinline__ void async_tile_load(float* ldsTile,
                                                const float* gbase,
                                                size_t elemOff, int lane) {
    unsigned ldsAddr = (unsigned)(uintptr_t)ldsTile + (unsigned)lane * 16u;
    unsigned gOff    = (unsigned)(elemOff * 4u) + (unsigned)lane * 16u;
    asm volatile(
        "global_load_async_to_lds_b128 %0, %1, %2 offset:0\n\t"
        "global_load_async_to_lds_b128 %0, %1, %2 offset:512"
        :: "v"(ldsAddr), "v"(gOff), "s"(gbase) : "memory");
}

// C = A @ B^T (== A @ B for the symmetric B operands used throughout).
// 16x16x16 via 4x V_WMMA_F32_16X16X4_F32; operands are row-major LDS tiles.
// Safe to call with lC aliasing lA/lB: operand DS loads issue before C DS
// stores in program order and LDS is in-order per wave.
__device__ __forceinline__ void mm16(const float* lA, const float* lB,
                                     float* lC, int lane) {
    const int half = lane >> 4;     // 0: lanes 0-15, 1: lanes 16-31
    const int lid  = lane & 15;
    const int kb   = half * 2;      // K base within a K=4 slice

    v2f a[4], b[4];
#pragma unroll
    for (int kk = 0; kk < 4; ++kk) {
        const int k0 = kk * 4 + kb;
        // A 16x4 slice: lanes 0-15 hold K={0,1}, lanes 16-31 hold K={2,3}
        a[kk].x = lA[lid * 16 + k0];
        a[kk].y = lA[lid * 16 + k0 + 1];
        // B read transposed (symmetric): contiguous pair -> b64-mergeable
        b[kk].x = lB[lid * 16 + k0];
        b[kk].y = lB[lid * 16 + k0 + 1];
    }

    v8f c = {};
#pragma unroll
    for (int kk = 0; kk < 4; ++kk) {
        // (neg_a, A, neg_b, B, c_mod, C, reuse_a, reuse_b)
        c = __builtin_amdgcn_wmma_f32_16x16x4_f32(
                false, a[kk], false, b[kk], (short)0, c, false, false);
    }

#pragma unroll
    for (int r = 0; r < 8; ++r)
        lC[(r + half * 8) * 16 + lid] = c[r];
    lds_fence();
}

__global__ __launch_bounds__(NW * 32, 1)
void leapfrog_spd_kernel(const float* __restrict__ mu,
                         const float* __restrict__ Sigma,
                         const float* __restrict__ phi,
                         const float* __restrict__ pi_mu,
                         const float* __restrict__ pi_Sigma,
                         const float* __restrict__ pi_phi,
                         const float* __restrict__ M_inv,
                         const float* __restrict__ gV_mu,
                         const float* __restrict__ gV_Sigma,
                         const float* __restrict__ gV_phi,
                         float* __restrict__ out) {
    __shared__ float lds[NW * NTILES * TILE];   // 36 KB / block

    const int lane = threadIdx.x & 31;
    const int wv   = threadIdx.x >> 5;
    const int mi   = blockIdx.x * NW + wv;      // matrix index
    if (mi >= NMAT) return;                     // wave-uniform

    float* tl = &lds[wv * NTILES * TILE];
    float* S  = tl + 0 * TILE;   // Sigma (later Sigma_new)
    float* P  = tl + 1 * TILE;   // pi_Sigma (later pi_Sigma_h / pi_Sigma_new)
    float* G  = tl + 2 * TILE;   // grad_V_Sigma
    float* T1 = tl + 3 * TILE;
    float* T2 = tl + 4 * TILE;
    float* Vt = tl + 5 * TILE;   // tangent V, then scaled W
    float* Yt = tl + 6 * TILE;   // NS: Y -> sqrt(S/c)
    float* Zt = tl + 7 * TILE;   // NS: Z -> (S/c)^{-1/2}
    float* E  = tl + 8 * TILE;   // expm accumulator

    const size_t mb = (size_t)mi * TILE;

    // ---- async global -> LDS staging of the three input tiles ------------
    async_tile_load(S, Sigma,    mb, lane);
    async_tile_load(P, pi_Sigma, mb, lane);
    async_tile_load(G, gV_Sigma, mb, lane);
    async_fence();

    // ---- half kick: pi_Sigma_h = P - h*(G + 2*sym(P S P)) ----------------
    mm16(P, S, T1, lane);        // P @ S
    mm16(T1, P, T2, lane);       // (P S) @ P
#pragma unroll
    for (int i = 0; i < 8; ++i) {
        const int idx = i * 32 + lane;
        P[idx] = P[idx] - H_C * (G[idx] + T2[idx] + T2[trI(idx)]);
    }
    lds_fence();

    // ---- drift tangent: V = dt * sym(2 S Ph S) = dt*(T2 + T2^T) ----------
    mm16(S, P, T1, lane);        // S @ Ph
    mm16(T1, S, T2, lane);       // (S Ph) @ S
#pragma unroll
    for (int i = 0; i < 8; ++i) {
        const int idx = i * 32 + lane;
        Vt[idx] = DT_C * (T2[idx] + T2[trI(idx)]);
    }
    lds_fence();

    // ---- Frobenius norm of Sigma (wave reduction) ------------------------
    float ss = 0.f;
#pragma unroll
    for (int i = 0; i < 8; ++i) {
        const float v = S[i * 32 + lane];
        ss += v * v;
    }
#pragma unroll
    for (int off = 16; off > 0; off >>= 1)
        ss += __shfl_xor(ss, off, 32);
    const float cN  = sqrtf(ss) + 1e-12f;   // eig(S/cN) in (0,1]
    const float rcN = 1.0f / cN;

    // ---- Newton-Schulz coupled iteration: Y->sqrt(S/c), Z->(S/c)^{-1/2} --
#pragma unroll
    for (int i = 0; i < 8; ++i) {
        const int idx = i * 32 + lane;
        Yt[idx] = S[idx] * rcN;
        Zt[idx] = (idx % 17 == 0) ? 1.0f : 0.0f;
    }
    lds_fence();

    for (int it = 0; it < 12; ++it) {
        mm16(Zt, Yt, T1, lane);                  // Z @ Y
#pragma unroll
        for (int i = 0; i < 8; ++i) {
            const int idx = i * 32 + lane;
            T1[idx] = ((idx % 17 == 0) ? 1.5f : 0.0f) - 0.5f * T1[idx];
        }
        lds_fence();
        mm16(Yt, T1, Yt, lane);                  // Y = Y @ T   (in-place OK)
        mm16(T1, Zt, Zt, lane);                  // Z = T @ Z   (in-place OK)
    }
    // S_half = sqrt(c)*Y ; S_ihalf = Z/sqrt(c)

    // ---- W_s = sym(S_ihalf V S_ihalf) * 2^-4 = (1/c)*sym(Z V Z)/16 -------
    mm16(Zt, Vt, T1, lane);
    mm16(T1, Zt, T2, lane);
    const float wscale = 0.5f * rcN * (1.0f / 16.0f);
#pragma unroll
    for (int i = 0; i < 8; ++i) {
        const int idx = i * 32 + lane;
        Vt[idx] = wscale * (T2[idx] + T2[trI(idx)]);
    }
    lds_fence();

    // ---- expm(W_s): order-3 Taylor (||W_s|| ~ 6e-4), then 4 squarings ----
    mm16(Vt, Vt, T1, lane);                      // W^2
    mm16(T1, Vt, T2, lane);                      // W^3
#pragma unroll
    for (int i = 0; i < 8; ++i) {
        const int idx = i * 32 + lane;
        E[idx] = ((idx % 17 == 0) ? 1.0f : 0.0f)
               + Vt[idx] + 0.5f * T1[idx] + (1.0f / 6.0f) * T2[idx];
    }
    lds_fence();
    for (int s = 0; s < 4; ++s)
        mm16(E, E, E, lane);                     // E = E^2 (in-place OK)

    // ---- Sigma_new = c * sym(Y E Y) --------------------------------------
    mm16(Yt, E, T1, lane);
    mm16(T1, Yt, T2, lane);
#pragma unroll
    for (int i = 0; i < 8; ++i) {
        const int idx = i * 32 + lane;
        S[idx] = cN * 0.5f * (T2[idx] + T2[trI(idx)]);
    }
    lds_fence();

    // ---- second half kick with frozen potential force --------------------
    mm16(P, S, T1, lane);        // Ph @ Sigma_new
    mm16(T1, P, T2, lane);       // (Ph Snew) @ Ph
#pragma unroll
    for (int i = 0; i < 8; ++i) {
        const int idx = i * 32 + lane;
        P[idx] = P[idx] - H_C * (G[idx] + T2[idx] + T2[trI(idx)]);
    }
    lds_fence();

    // ---- write matrix outputs (coalesced b128) ---------------------------
    float4* oS = (float4*)(out + OFF_SIGMA + mb);
    float4* oP = (float4*)(out + OFF_PISIG + mb);
    const float4* s4 = (const float4*)S;
    const float4* p4 = (const float4*)P;
    oS[lane]      = s4[lane];
    oS[lane + 32] = s4[lane + 32];
    oP[lane]      = p4[lane];
    oP[lane + 32] = p4[lane + 32];

    // ---- mu / pi_mu: matvec with M_inv (lanes 0-15) ---------------------
    const size_t vb = (size_t)mi * KDIM;
    if (lane < KDIM) {
        T1[lane] = pi_mu[vb + lane] - H_C * gV_mu[vb + lane];  // pi_mu_h
    }
    lds_fence();
    if (lane < KDIM) {
        float acc = 0.f;
#pragma unroll
        for (int j = 0; j < KDIM; ++j)
            acc += M_inv[mb + lane * KDIM + j] * T1[j];
        out[OFF_MU   + vb + lane] = mu[vb + lane] + DT_C * acc;
        out[OFF_PIMU + vb + lane] = T1[lane] - H_C * gV_mu[vb + lane];
    }

    // ---- phi / pi_phi (lanes 0-2) ---------------------------------------
    const size_t pb = (size_t)mi * 3;
    if (lane < 3) {
        const float gp  = gV_phi[pb + lane];
        const float pph = pi_phi[pb + lane] - H_C * gp;        // pi_phi_h
        out[OFF_PHI   + pb + lane] = phi[pb + lane] + DT_C * pph;
        out[OFF_PIPHI + pb + lane] = pph - H_C * gp;
    }
}

extern "C" void kernel_launch(void* const* d_in, const int* in_sizes, int n_in,
                              void* d_out, int out_size, void* d_ws, size_t ws_size,
                              hipStream_t stream) {
    const float* mu       = (const float*)d_in[0];
    const float* Sigma    = (const float*)d_in[1];
    const float* phi      = (const float*)d_in[2];
    const float* pi_mu    = (const float*)d_in[3];
    const float* pi_Sigma = (const float*)d_in[4];
    const float* pi_phi   = (const float*)d_in[5];
    const float* M_inv    = (const float*)d_in[6];
    const float* gV_mu    = (const float*)d_in[7];
    const float* gV_Sigma = (const float*)d_in[8];
    const float* gV_phi   = (const float*)d_in[9];

    dim3 block(NW * 32);
    dim3 grid((NMAT + NW - 1) / NW);   // 8192 blocks, one wave per matrix
    leapfrog_spd_kernel<<<grid, block, 0, stream>>>(
        mu, Sigma, phi, pi_mu, pi_Sigma, pi_phi,
        M_inv, gV_mu, gV_Sigma, gV_phi, (float*)d_out);
}